// Backwarp_62680752718019
// MI455X (gfx1250) — compile-verified
//
#include <hip/hip_runtime.h>

// Dense image warp (bilinear backward warp), B=4, H=512, W=512, C=64, fp32.
// Memory-bound gather: ~545 MB of HBM traffic, ~0.54 GFLOP -> pure bandwidth.
// Layout: 16 lanes per pixel, one float4 channel-quad per lane (C=64 = 16*4).

typedef float v4f __attribute__((ext_vector_type(4)));
typedef float v2f __attribute__((ext_vector_type(2)));

namespace {
constexpr int kB = 4;
constexpr int kH = 512;
constexpr int kW = 512;
constexpr int kC = 64;
constexpr int kRowStride = kW * kC;           // floats per image row
constexpr int kLanesPerPixel = kC / 4;        // 16
constexpr int kTotalThreads = kB * kH * kW * kLanesPerPixel;  // 16,777,216
constexpr int kBlock = 256;
}  // namespace

__global__ __launch_bounds__(kBlock) void dense_warp_kernel(
    const float* __restrict__ image,
    const float* __restrict__ flow,
    float* __restrict__ out) {
  const int gid = blockIdx.x * kBlock + threadIdx.x;
  const int pix = gid >> 4;        // pixel index in [0, B*H*W)
  const int cq  = (gid & 15) << 2; // channel quad offset: 0,4,...,60

  // H*W = 2^18, W = 2^9 -> pure shifts/masks, no integer division.
  const int b = pix >> 18;
  const int y = (pix >> 9) & (kW - 1);
  const int x = pix & (kW - 1);

  // Per-pixel flow; 16 lanes of a pixel read the same 8B -> merged in cache.
  const v2f f = *reinterpret_cast<const v2f*>(flow + (size_t)pix * 2);

  const float qy = (float)y - f.x;
  const float qx = (float)x - f.y;

  float fy = floorf(qy);
  fy = fminf(fmaxf(fy, 0.0f), (float)(kH - 2));
  float fx = floorf(qx);
  fx = fminf(fmaxf(fx, 0.0f), (float)(kW - 2));
  const float ay = fminf(fmaxf(qy - fy, 0.0f), 1.0f);
  const float ax = fminf(fmaxf(qx - fx, 0.0f), 1.0f);
  const int y0 = (int)fy;
  const int x0 = (int)fx;

  // Top-left corner base for this channel quad (16B aligned: cq % 4 == 0).
  const float* p00 =
      image + ((size_t)((b * kH + y0) * kW + x0) * kC + cq);

  const v4f tl = *reinterpret_cast<const v4f*>(p00);
  const v4f tr = *reinterpret_cast<const v4f*>(p00 + kC);
  const v4f bl = *reinterpret_cast<const v4f*>(p00 + kRowStride);
  const v4f br = *reinterpret_cast<const v4f*>(p00 + kRowStride + kC);

  // top = tl + ax*(tr-tl); bottom = bl + ax*(br-bl); out = top + ay*(bottom-top)
  v4f top, bot, res;
  top.x = fmaf(ax, tr.x - tl.x, tl.x);
  top.y = fmaf(ax, tr.y - tl.y, tl.y);
  top.z = fmaf(ax, tr.z - tl.z, tl.z);
  top.w = fmaf(ax, tr.w - tl.w, tl.w);
  bot.x = fmaf(ax, br.x - bl.x, bl.x);
  bot.y = fmaf(ax, br.y - bl.y, bl.y);
  bot.z = fmaf(ax, br.z - bl.z, bl.z);
  bot.w = fmaf(ax, br.w - bl.w, bl.w);
  res.x = fmaf(ay, bot.x - top.x, top.x);
  res.y = fmaf(ay, bot.y - top.y, top.y);
  res.z = fmaf(ay, bot.z - top.z, top.z);
  res.w = fmaf(ay, bot.w - top.w, top.w);

  // Streaming 268MB output: non-temporal store keeps the image resident in
  // the 192MB L2 instead of being evicted by write data.
  v4f* dst = reinterpret_cast<v4f*>(out + (size_t)pix * kC + cq);
  __builtin_nontemporal_store(res, dst);
}

extern "C" void kernel_launch(void* const* d_in, const int* in_sizes, int n_in,
                              void* d_out, int out_size, void* d_ws, size_t ws_size,
                              hipStream_t stream) {
  const float* image = (const float*)d_in[0];  // [B,H,W,C] fp32
  const float* flow  = (const float*)d_in[1];  // [B,H,W,2] fp32
  float* out = (float*)d_out;                  // [B,H,W,C] fp32

  const int grid = kTotalThreads / kBlock;     // 65536 blocks
  dense_warp_kernel<<<grid, kBlock, 0, stream>>>(image, flow, out);
}